// LaplacianLoss_59373627900186
// MI455X (gfx1250) — compile-verified
//
#include <hip/hip_runtime.h>
#include <hip/hip_bf16.h>
#include <stdint.h>

// ---------------------------------------------------------------------------
// LaplacianLoss on MI455X (gfx1250, wave32)
//   loss = mean_{b,n,c} | d[b,n,c] - nsum[b,n,c]/deg[n] |   (0 where deg==0)
//   d = pred - target,  nsum[dst] += d[src] over edges,  deg[n] = in-degree
// ---------------------------------------------------------------------------

typedef __attribute__((ext_vector_type(2))) float v2f;
typedef __attribute__((ext_vector_type(8))) float v8f;

// ---- CDNA5 async global->LDS staging (builtin verified present) ------------
__device__ __forceinline__ void async_ld_i32(const int* g, int* l) {
#if __has_builtin(__builtin_amdgcn_global_load_async_to_lds_b32)
  typedef __attribute__((address_space(1))) int gint;
  typedef __attribute__((address_space(3))) int lint;
  __builtin_amdgcn_global_load_async_to_lds_b32((gint*)g, (lint*)l, 0, 0);
#else
  *l = *g;
#endif
}

__device__ __forceinline__ void wait_async0() {
#if __has_builtin(__builtin_amdgcn_s_wait_asynccnt)
  __builtin_amdgcn_s_wait_asynccnt(0);
#elif defined(__gfx1250__)
  asm volatile("s_wait_asynccnt 0" ::: "memory");
#endif
}

// ---- 1) zero workspace ----------------------------------------------------
__global__ void lap_zero_kernel(unsigned int* ws, long long words) {
  long long i = (long long)blockIdx.x * blockDim.x + threadIdx.x;
  long long stride = (long long)gridDim.x * blockDim.x;
  for (; i < words; i += stride) ws[i] = 0u;
}

// ---- 2) in-degree ----------------------------------------------------------
__global__ void lap_degree_kernel(const int* __restrict__ edst,
                                  int* __restrict__ deg, int E) {
  int i = blockIdx.x * blockDim.x + threadIdx.x;
  int stride = gridDim.x * blockDim.x;
  for (; i < E; i += stride) atomicAdd(&deg[edst[i]], 1);
}

// ---- 3) scatter: nsum[dst] += (pred-target)[src] ---------------------------
// Block = 256 threads = 8 waves. Each block stages a 256-edge chunk of
// (src,dst) into LDS via async global->LDS loads, then each wave processes
// edges with lane == channel c (C=32): per batch slice a fully coalesced
// 128B gather of d[src] and 128B of f32 atomic adds to nsum[dst].
#define LAP_CHUNK 256
__global__ void lap_scatter_kernel(const float* __restrict__ pred,
                                   const float* __restrict__ target,
                                   const int* __restrict__ esrc,
                                   const int* __restrict__ edst,
                                   float* __restrict__ nsum,
                                   int N, int E) {
  __shared__ int s_src[LAP_CHUNK];
  __shared__ int s_dst[LAP_CHUNK];
  const int tid  = threadIdx.x;
  const int lane = tid & 31;      // channel c
  const int wave = tid >> 5;      // 0..7

  for (int base = blockIdx.x * LAP_CHUNK; base < E;
       base += gridDim.x * LAP_CHUNK) {
    const int n = min(LAP_CHUNK, E - base);
    if (tid < n) {
      async_ld_i32(&esrc[base + tid], &s_src[tid]);
      async_ld_i32(&edst[base + tid], &s_dst[tid]);
    }
    wait_async0();
    __syncthreads();

    for (int e = wave; e < n; e += 8) {
      const int src = s_src[e];
      const int dst = s_dst[e];
#pragma unroll
      for (int b = 0; b < 4; ++b) {
        const size_t si = ((size_t)b * N + src) * 32 + lane;
        const size_t di = ((size_t)b * N + dst) * 32 + lane;
        const float d = pred[si] - target[si];
        atomicAdd(&nsum[di], d);
      }
    }
    __syncthreads();
  }
}

// ---- 4) abs-Laplacian reduction, wave-per-node, WMMA cross-lane sum --------
// lane == channel c (C==32), loop over B==4 slices: coalesced 128B loads,
// deg[node] read once per wave, one reciprocal per node (no per-element div,
// no 64-bit modulo). Node loop is wave-uniform -> EXEC all-1s at the WMMA.
//
// Wave reduction via v_wmma_f32_16x16x4_f32:
//   A[lane][0] = acc (lanes 0..15 -> rows, K=0; lanes 16..31 -> K=2)
//   B = ones(4x16)  =>  D[m][n] = acc_m + acc_{m+16} for every n.
// Lane0's 8 C-VGPRs hold rows 0..7, lane16's hold rows 8..15.
__global__ void lap_loss_kernel(const float* __restrict__ pred,
                                const float* __restrict__ target,
                                const float* __restrict__ nsum,
                                const int* __restrict__ deg,
                                double* __restrict__ acc, int N) {
  const int lane   = threadIdx.x & 31;
  const int wib    = threadIdx.x >> 5;               // wave in block
  const int wpb    = blockDim.x >> 5;                // waves per block
  const int wav    = blockIdx.x * wpb + wib;         // global wave id
  const int nwaves = gridDim.x * wpb;

  float a = 0.0f;
  for (int node = wav; node < N; node += nwaves) {
    const int dg = deg[node];
    const float invdeg = (dg > 0) ? (1.0f / (float)dg) : 0.0f;
#pragma unroll
    for (int b = 0; b < 4; ++b) {
      const size_t i = ((size_t)b * N + node) * 32 + lane;
      const float d = pred[i] - target[i];
      a += (dg > 0) ? fabsf(d - nsum[i] * invdeg) : 0.0f;
    }
  }

  float tot;
#if __has_builtin(__builtin_amdgcn_wmma_f32_16x16x4_f32)
  v2f A; A.x = a;    A.y = 0.0f;
  v2f Bv; Bv.x = 1.0f; Bv.y = 1.0f;           // ones matrix
  v8f Cc = {};
  Cc = __builtin_amdgcn_wmma_f32_16x16x4_f32(
      /*neg_a=*/false, A, /*neg_b=*/false, Bv,
      /*c_mod=*/(short)0, Cc, /*reuse_a=*/false, /*reuse_b=*/false);
  float s8 = Cc[0] + Cc[1] + Cc[2] + Cc[3] + Cc[4] + Cc[5] + Cc[6] + Cc[7];
  tot = s8 + __shfl_down(s8, 16, 32);         // rows 0..7(+16..23) + rows 8..15
#else
  tot = a;
  for (int off = 16; off > 0; off >>= 1) tot += __shfl_down(tot, off, 32);
#endif

  if (lane == 0) atomicAdd(acc, (double)tot);
}

// ---- 5) finalize -----------------------------------------------------------
__global__ void lap_finalize_kernel(const double* __restrict__ acc,
                                    float* __restrict__ out, double inv_total) {
  if (blockIdx.x == 0 && threadIdx.x == 0) out[0] = (float)(acc[0] * inv_total);
}

// ---------------------------------------------------------------------------
extern "C" void kernel_launch(void* const* d_in, const int* in_sizes, int n_in,
                              void* d_out, int out_size, void* d_ws, size_t ws_size,
                              hipStream_t stream) {
  const float* pred   = (const float*)d_in[0];
  const float* target = (const float*)d_in[1];
  const int*   esrc   = (const int*)d_in[2];
  const int*   edst   = (const int*)d_in[3];

  const long long BNC = (long long)in_sizes[0];   // B*N*C = 12.8M
  const int E = in_sizes[2];
  const int B = 4, C = 32;
  const int N = (int)(BNC / ((long long)B * C));  // 100000

  // workspace layout: nsum f32[BNC] | deg i32[N] | acc f64 (8B aligned)
  float* nsum = (float*)d_ws;
  int*   deg  = (int*)((char*)d_ws + sizeof(float) * BNC);
  size_t accOff = sizeof(float) * BNC + sizeof(int) * (size_t)N;
  accOff = (accOff + 7) & ~(size_t)7;
  double* acc = (double*)((char*)d_ws + accOff);

  const long long zeroWords = (long long)(accOff / 4) + 2;   // nsum+deg+acc
  lap_zero_kernel<<<2048, 256, 0, stream>>>((unsigned int*)d_ws, zeroWords);
  lap_degree_kernel<<<2048, 256, 0, stream>>>(edst, deg, E);
  lap_scatter_kernel<<<2048, 256, 0, stream>>>(pred, target, esrc, edst,
                                               nsum, N, E);
  lap_loss_kernel<<<1024, 256, 0, stream>>>(pred, target, nsum, deg, acc, N);
  lap_finalize_kernel<<<1, 32, 0, stream>>>(acc, (float*)d_out,
                                            1.0 / (double)BNC);
}